// LocalEdgeConv_81870666596741
// MI455X (gfx1250) — compile-verified
//
#include <hip/hip_runtime.h>
#include <hip/hip_bf16.h>

// ---------------------------------------------------------------------------
// DGCNN edge-conv block for MI455X (gfx1250), fp32 via V_WMMA_F32_16X16X4_F32.
// Pipeline: xx -> knn(WMMA gram + reg top-20) -> conv1(WMMA, LDS edge feats)
//           -> BN1 finalize -> conv2(WMMA, in-place) -> BN2 finalize -> maxpool.
// All matrix math uses wave32 WMMA f32 16x16x4 (exact-precision path; kNN
// ranking is precision sensitive so fp32 is the safe choice, and GEMM FLOPs
// (~17 GF) are tiny vs. the chip's matrix throughput).
// ---------------------------------------------------------------------------

typedef __attribute__((ext_vector_type(2))) float v2f;
typedef __attribute__((ext_vector_type(8))) float v8f;

// Only check on the DEVICE pass; the builtin is (correctly) absent on host.
#if defined(__HIP_DEVICE_COMPILE__)
#if !__has_builtin(__builtin_amdgcn_wmma_f32_16x16x4_f32)
#error "gfx1250 device pass: __builtin_amdgcn_wmma_f32_16x16x4_f32 missing"
#endif
#endif

// D = A(16x4 f32) * B(4x16 f32) + C(16x16 f32)
// args: (neg_a, A, neg_b, B, c_mod, C, reuse_a, reuse_b)
static __device__ inline v8f wmma_f32x4(v2f a, v2f b, v8f c) {
#if defined(__HIP_DEVICE_COMPILE__)
  return __builtin_amdgcn_wmma_f32_16x16x4_f32(false, a, false, b, (short)0, c,
                                               false, false);
#else
  return c;  // host pass never executes this
#endif
}

// Problem constants (match setup_inputs()).
constexpr int Bc = 4;
constexpr int Cc = 64;     // input channels
constexpr int Nc = 4096;   // points
constexpr int Oc = 64;     // output channels
constexpr int Kc = 20;     // neighbours
constexpr int C2 = 128;    // 2*Cc edge-feature channels
constexpr int PTS = 4;     // points per conv workgroup
constexpr int MROWS = PTS * Kc;  // 80 edge rows per conv workgroup

// ---------------------------------------------------------------------------
// Squared norms xx[b][n] = sum_c x[b][c][n]^2
// ---------------------------------------------------------------------------
__global__ __launch_bounds__(256) void xx_kernel(const float* __restrict__ x,
                                                 float* __restrict__ xx) {
  int i = blockIdx.x * 256 + threadIdx.x;  // over B*N
  int b = i >> 12;                         // N = 4096 = 2^12
  int n = i & (Nc - 1);
  const float* xb = x + (size_t)b * Cc * Nc;
  float s = 0.f;
#pragma unroll 8
  for (int c = 0; c < Cc; ++c) {
    float v = xb[(size_t)c * Nc + n];
    s += v * v;
  }
  xx[i] = s;
}

// ---------------------------------------------------------------------------
// kNN: one workgroup = 16 query rows of one batch. 8 waves each produce a
// 16x16 neg-dist tile per outer step (128 columns/step) via WMMA f32 16x16x4.
// Top-20 kept per (row, sub-thread) in registers with unrolled bubble
// insertion (data-independent indices -> stays in VGPRs), merged via LDS.
// ---------------------------------------------------------------------------
__global__ __launch_bounds__(256) void knn_kernel(const float* __restrict__ x,
                                                  const float* __restrict__ xx,
                                                  int* __restrict__ idx_out) {
  __shared__ float sA[Cc][16];          // A block: 16 rows x 64 ch (c-major), 4KB
  __shared__ float sXXi[16];
  __shared__ float sD[8][16][16];       // per-wave neg-dist tiles, 8KB
  __shared__ float sMV[16][16 * Kc];    // merge candidates (vals), 20KB
  __shared__ int sMI[16][16 * Kc];      // merge candidates (idx), 20KB

  const int b = blockIdx.y;
  const int i0 = blockIdx.x * 16;
  const int tid = threadIdx.x;
  const int wave = tid >> 5;
  const int lane = tid & 31;
  const float* xb = x + (size_t)b * Cc * Nc;

  // Stage the 16-row A block + its norms.
  for (int e = tid; e < Cc * 16; e += 256) {
    int c = e >> 4, m = e & 15;
    sA[c][m] = xb[(size_t)c * Nc + i0 + m];
  }
  if (tid < 16) sXXi[tid] = xx[b * Nc + i0 + tid];
  __syncthreads();

  // Register-resident top-20 (descending by neg-dist = nearest first).
  float tv[Kc];
  int ti[Kc];
#pragma unroll
  for (int t = 0; t < Kc; ++t) {
    tv[t] = -3.0e38f;
    ti[t] = 0;
  }
  const int row = tid >> 4;  // 0..15 (query row)
  const int sub = tid & 15;  // 16 sub-threads per row

  const int m = lane & 15;     // row (A) / col (B) within tile
  const int khalf = lane >> 4; // 0: K={0,1}, 1: K={2,3} per step

  for (int j0 = 0; j0 < Nc; j0 += 128) {
    const int jw = j0 + wave * 16;  // this wave's 16 columns
    v8f acc = {};
    const float* xcol = xb + (jw + m);
#pragma unroll
    for (int kk = 0; kk < Cc / 4; ++kk) {  // 16 WMMA steps over K=64
      const int k0 = kk * 4 + khalf * 2;
      v2f a, bf;
      a.x = sA[k0][m];
      a.y = sA[k0 + 1][m];
      bf.x = xcol[(size_t)k0 * Nc];
      bf.y = xcol[(size_t)(k0 + 1) * Nc];
      acc = wmma_f32x4(a, bf, acc);
    }
    // neg_dist = 2*inner - ||xi||^2 - ||xj||^2 ; write tile to LDS.
    const float xxj = xx[b * Nc + jw + m];
#pragma unroll
    for (int r = 0; r < 8; ++r) {
      const int mm = r + khalf * 8;
      sD[wave][mm][m] = 2.0f * acc[r] - sXXi[mm] - xxj;
    }
    __syncthreads();
    // Top-k update: thread (row, sub) consumes col = j0 + w*16 + sub.
#pragma unroll
    for (int w = 0; w < 8; ++w) {
      float v = sD[w][row][sub];
      int ci = j0 + w * 16 + sub;
      if (v > tv[Kc - 1]) {
        float cv = v;
#pragma unroll
        for (int t = 0; t < Kc; ++t) {
          if (cv > tv[t]) {
            float fv = tv[t];
            int fi = ti[t];
            tv[t] = cv;
            ti[t] = ci;
            cv = fv;
            ci = fi;
          }
        }
      }
    }
    __syncthreads();
  }

  // Merge the 16 partial lists per row.
#pragma unroll
  for (int t = 0; t < Kc; ++t) {
    sMV[row][sub * Kc + t] = tv[t];
    sMI[row][sub * Kc + t] = ti[t];
  }
  __syncthreads();
  if (tid < 16) {
    const int r = tid;
    float bv[Kc];
    int bi[Kc];
#pragma unroll
    for (int t = 0; t < Kc; ++t) {
      bv[t] = -3.0e38f;
      bi[t] = 0;
    }
    for (int e = 0; e < 16 * Kc; ++e) {
      float v = sMV[r][e];
      int ci = sMI[r][e];
      if (v > bv[Kc - 1]) {
        float cv = v;
#pragma unroll
        for (int t = 0; t < Kc; ++t) {
          if (cv > bv[t]) {
            float fv = bv[t];
            int fi = bi[t];
            bv[t] = cv;
            bi[t] = ci;
            cv = fv;
            ci = fi;
          }
        }
      }
    }
#pragma unroll
    for (int t = 0; t < Kc; ++t)
      idx_out[((size_t)b * Nc + i0 + r) * Kc + t] = bi[t];
  }
}

// ---------------------------------------------------------------------------
// conv1: build edge features for 4 points (80 rows x 128 ch) in LDS, then
// 5 M-tiles x 4 N-tiles of WMMA f32 16x16x4 over K=128 against w1.
// Accumulate per-channel sum/sumsq (ds_add_f32 then 1 global atomic/channel).
// ---------------------------------------------------------------------------
__global__ __launch_bounds__(256) void conv1_kernel(
    const float* __restrict__ x, const int* __restrict__ idx,
    const float* __restrict__ w1, const float* __restrict__ b1,
    float* __restrict__ y, float* __restrict__ stats) {
  __shared__ float sE[MROWS][C2];  // 40KB edge-feature tile
  __shared__ float sSum[Oc];
  __shared__ float sSq[Oc];

  const int b = blockIdx.y;
  const int p0 = blockIdx.x * PTS;
  const int tid = threadIdx.x;
  if (tid < Oc) {
    sSum[tid] = 0.f;
    sSq[tid] = 0.f;
  }
  const float* xb = x + (size_t)b * Cc * Nc;

  // Edge features: [neigh - center, center]. Never touches HBM as a tensor.
  for (int e = tid; e < MROWS * C2; e += 256) {
    const int rr = e >> 7;  // 0..79
    const int c = e & 127;
    const int p = rr / Kc;
    const int kk = rr - p * Kc;
    const int n = p0 + p;
    const int j = idx[((size_t)b * Nc + n) * Kc + kk];
    const float center = xb[(size_t)(c & 63) * Nc + n];
    sE[rr][c] = (c < Cc) ? (xb[(size_t)c * Nc + j] - center) : center;
  }
  __syncthreads();

  const int wave = tid >> 5, lane = tid & 31;
  const int m = lane & 15, khalf = lane >> 4;
  const size_t rbase = ((size_t)b * Nc + p0) * Kc;  // 80 contiguous rows

  for (int t = wave; t < 20; t += 8) {  // 5 M-tiles x 4 N-tiles
    const int mt = t >> 2, nt = t & 3;
    v8f acc = {};
#pragma unroll
    for (int kk = 0; kk < C2 / 4; ++kk) {  // 32 WMMA steps, K=128
      const int k0 = kk * 4 + khalf * 2;
      v2f a, bf;
      a.x = sE[mt * 16 + m][k0];
      a.y = sE[mt * 16 + m][k0 + 1];
      bf.x = w1[(nt * 16 + m) * C2 + k0];  // B[c][o] = w1[o][c]
      bf.y = w1[(nt * 16 + m) * C2 + k0 + 1];
      acc = wmma_f32x4(a, bf, acc);
    }
    const int o = nt * 16 + m;
    const float bias = b1[o];
    float lsum = 0.f, lsq = 0.f;
#pragma unroll
    for (int r = 0; r < 8; ++r) {
      const int mm = mt * 16 + r + khalf * 8;  // 0..79
      const float v = acc[r] + bias;
      y[(rbase + mm) * Oc + o] = v;
      lsum += v;
      lsq += v * v;
    }
    atomicAdd(&sSum[o], lsum);  // ds_add_f32
    atomicAdd(&sSq[o], lsq);
  }
  __syncthreads();
  if (tid < Oc) {
    atomicAdd(&stats[tid], sSum[tid]);
    atomicAdd(&stats[Oc + tid], sSq[tid]);
  }
}

// ---------------------------------------------------------------------------
// BN finalize: fold mean/var/gamma/beta -> per-channel (scale, shift).
// ---------------------------------------------------------------------------
__global__ void bn_finalize_kernel(const float* __restrict__ stats,
                                   const float* __restrict__ gamma,
                                   const float* __restrict__ beta,
                                   float* __restrict__ ss, float inv_count) {
  const int o = threadIdx.x;  // Oc threads
  const float mean = stats[o] * inv_count;
  const float var = stats[Oc + o] * inv_count - mean * mean;
  const float sc = gamma[o] * rsqrtf(var + 1e-5f);
  ss[o] = sc;
  ss[Oc + o] = beta[o] - mean * sc;
}

// ---------------------------------------------------------------------------
// conv2: BN1+ReLU applied while staging 80x64 rows into LDS, then WMMA over
// K=64 against w2, written back IN PLACE (reads complete before the barrier;
// each block owns its 80 rows). Accumulates BN2 stats.
// ---------------------------------------------------------------------------
__global__ __launch_bounds__(256) void conv2_kernel(
    const float* __restrict__ w2, const float* __restrict__ b2,
    const float* __restrict__ ss1, float* __restrict__ y,
    float* __restrict__ stats2) {
  __shared__ float sE[MROWS][Oc];  // 20KB
  __shared__ float sSum[Oc];
  __shared__ float sSq[Oc];

  const int b = blockIdx.y;
  const int p0 = blockIdx.x * PTS;
  const int tid = threadIdx.x;
  if (tid < Oc) {
    sSum[tid] = 0.f;
    sSq[tid] = 0.f;
  }
  const size_t rbase = ((size_t)b * Nc + p0) * Kc;

  for (int e = tid; e < MROWS * Oc; e += 256) {
    const int rr = e >> 6;
    const int c = e & 63;
    float v = y[(rbase + rr) * Oc + c];
    sE[rr][c] = fmaxf(v * ss1[c] + ss1[Oc + c], 0.f);  // BN1 + ReLU
  }
  __syncthreads();

  const int wave = tid >> 5, lane = tid & 31;
  const int m = lane & 15, khalf = lane >> 4;

  for (int t = wave; t < 20; t += 8) {
    const int mt = t >> 2, nt = t & 3;
    v8f acc = {};
#pragma unroll
    for (int kk = 0; kk < Oc / 4; ++kk) {  // 16 WMMA steps, K=64
      const int k0 = kk * 4 + khalf * 2;
      v2f a, bf;
      a.x = sE[mt * 16 + m][k0];
      a.y = sE[mt * 16 + m][k0 + 1];
      bf.x = w2[(nt * 16 + m) * Oc + k0];
      bf.y = w2[(nt * 16 + m) * Oc + k0 + 1];
      acc = wmma_f32x4(a, bf, acc);
    }
    const int o = nt * 16 + m;
    const float bias = b2[o];
    float lsum = 0.f, lsq = 0.f;
#pragma unroll
    for (int r = 0; r < 8; ++r) {
      const int mm = mt * 16 + r + khalf * 8;
      const float v = acc[r] + bias;
      y[(rbase + mm) * Oc + o] = v;
      lsum += v;
      lsq += v * v;
    }
    atomicAdd(&sSum[o], lsum);
    atomicAdd(&sSq[o], lsq);
  }
  __syncthreads();
  if (tid < Oc) {
    atomicAdd(&stats2[tid], sSum[tid]);
    atomicAdd(&stats2[Oc + tid], sSq[tid]);
  }
}

// ---------------------------------------------------------------------------
// BN2 + ReLU + max over k, transposed write [B,O,N].
// ---------------------------------------------------------------------------
__global__ __launch_bounds__(256) void maxpool_kernel(
    const float* __restrict__ y, const float* __restrict__ ss2,
    float* __restrict__ out) {
  const int gid = blockIdx.x * 256 + threadIdx.x;  // over B*N*O
  const int o = gid & 63;
  const int n = (gid >> 6) & (Nc - 1);
  const int b = gid >> 18;
  const float sc = ss2[o];
  const float sh = ss2[Oc + o];
  const size_t base = (((size_t)b * Nc + n) * Kc) * Oc + o;
  float mval = -3.0e38f;
#pragma unroll
  for (int kk = 0; kk < Kc; ++kk) {
    const float v = fmaxf(y[base + (size_t)kk * Oc] * sc + sh, 0.f);
    mval = fmaxf(mval, v);
  }
  out[((size_t)b * Oc + o) * Nc + n] = mval;
}

__global__ void zero_stats_kernel(float* __restrict__ s) {
  s[threadIdx.x] = 0.f;  // 4*Oc = 256 floats (stats1 + stats2)
}

// ---------------------------------------------------------------------------
extern "C" void kernel_launch(void* const* d_in, const int* in_sizes, int n_in,
                              void* d_out, int out_size, void* d_ws,
                              size_t ws_size, hipStream_t stream) {
  const float* x = (const float*)d_in[0];
  const float* w1 = (const float*)d_in[1];
  const float* b1 = (const float*)d_in[2];
  const float* g1 = (const float*)d_in[3];
  const float* be1 = (const float*)d_in[4];
  const float* w2 = (const float*)d_in[5];
  const float* b2 = (const float*)d_in[6];
  const float* g2 = (const float*)d_in[7];
  const float* be2 = (const float*)d_in[8];
  float* out = (float*)d_out;

  // Workspace layout (floats / 4B ints, ~85.3 MB total):
  float* ws = (float*)d_ws;
  float* xx = ws;                                   // B*N
  int* idx = (int*)(ws + Bc * Nc);                  // B*N*Kc ints
  float* stats1 = ws + Bc * Nc + Bc * Nc * Kc;      // 2*Oc
  float* stats2 = stats1 + 2 * Oc;                  // 2*Oc
  float* ss1 = stats2 + 2 * Oc;                     // 2*Oc
  float* ss2 = ss1 + 2 * Oc;                        // 2*Oc
  float* y = ss2 + 2 * Oc;                          // B*N*Kc*Oc (84 MB)

  zero_stats_kernel<<<1, 4 * Oc, 0, stream>>>(stats1);
  xx_kernel<<<(Bc * Nc) / 256, 256, 0, stream>>>(x, xx);

  dim3 gknn(Nc / 16, Bc);
  knn_kernel<<<gknn, 256, 0, stream>>>(x, xx, idx);

  dim3 gconv(Nc / PTS, Bc);
  conv1_kernel<<<gconv, 256, 0, stream>>>(x, idx, w1, b1, y, stats1);

  const float inv_count = 1.0f / (float)(Bc * Nc * Kc);
  bn_finalize_kernel<<<1, Oc, 0, stream>>>(stats1, g1, be1, ss1, inv_count);

  conv2_kernel<<<gconv, 256, 0, stream>>>(w2, b2, ss1, y, stats2);
  bn_finalize_kernel<<<1, Oc, 0, stream>>>(stats2, g2, be2, ss2, inv_count);

  maxpool_kernel<<<(Bc * Nc * Oc) / 256, 256, 0, stream>>>(y, ss2, out);
}